// SA_loss_40355512714151
// MI455X (gfx1250) — compile-verified
//
#include <hip/hip_runtime.h>
#include <math.h>

// ---------------------------------------------------------------------------
// PAN-style discriminative embedding loss, CDNA5 (gfx1250, wave32).
// Memory-bound streaming reduction (~170 MB ≈ 7.3 us @ 23.3 TB/s).
// LDS-side accumulation (ds_add_f32 / ds_min_i32) minimizes global atomics;
// the 8x8 label-mean Gram matrix is computed with one V_WMMA_F32_16X16X4_F32.
// ---------------------------------------------------------------------------

typedef __attribute__((ext_vector_type(2))) float v2f;
typedef __attribute__((ext_vector_type(8))) float v8f;

#define Bn   8
#define Dch  4
#define Hn   640
#define Wn   640
#define HWn  (Hn * Wn)
#define Ln   8
#define NB   128            // blocks per image for pixel passes
#define CH   (HWn / NB)     // 3200 pixels per block

#define DVc  0.5f
#define DDc  1.5f

// workspace layout (units of 4 bytes)
#define OFF_SUMK   0                      // B*L*D = 256 floats (kernel-region emb sums)
#define OFF_CNTK   (OFF_SUMK + Bn*Ln*Dch) // 64 floats
#define OFF_CNTF   (OFF_CNTK + Bn*Ln)     // 64 floats
#define OFF_SUMV   (OFF_CNTF + Bn*Ln)     // 64 floats (l_agg numerators)
#define OFF_MEAN   (OFF_SUMV + Bn*Ln)     // 256 floats
#define OFF_CD     (OFF_MEAN + Bn*Ln*Dch) // 64 floats (coef_diag)
#define OFF_PER    (OFF_CD + Bn*Ln)       // 8 floats (l_dis + l_reg per image)
#define OFF_FIRST  (OFF_PER + Bn)         // 64 ints
#define OFF_SECOND (OFF_FIRST + Bn*Ln)    // 64 ints
#define WS_WORDS   (OFF_SECOND + Bn*Ln)   // 904 words

// ---------------------------------------------------------------------------
__global__ __launch_bounds__(256) void k_init(float* F, int* I) {
    int t = blockIdx.x * blockDim.x + threadIdx.x;
    if (t < OFF_FIRST)        F[t] = 0.0f;
    else if (t < WS_WORDS)    I[t] = HWn;      // sentinel for first/second
}

// ---------------------------------------------------------------------------
// Pass 1: per-label counts (full + kernel region), kernel-region emb sums,
// first occurrence (row-major min index). One coalesced read of everything.
__global__ __launch_bounds__(256) void k_stats(const float* __restrict__ emb,
                                               const int*   __restrict__ inst,
                                               const float* __restrict__ kern,
                                               const float* __restrict__ tmask,
                                               float* F, int* I) {
    const int b = blockIdx.x / NB, chunk = blockIdx.x % NB;
    const int tid = threadIdx.x;
    __shared__ float sSum[Ln * Dch];
    __shared__ float sCk[Ln];
    __shared__ float sCf[Ln];
    __shared__ int   sF[Ln];
    if (tid < Ln * Dch) sSum[tid] = 0.0f;
    if (tid < Ln) { sCk[tid] = 0.0f; sCf[tid] = 0.0f; sF[tid] = HWn; }
    __syncthreads();

    const int    base = b * HWn;
    const float* e0   = emb + (size_t)b * Dch * HWn;
    const int    start = chunk * CH;
    for (int p = start + tid; p < start + CH; p += 256) {
        __builtin_prefetch(&inst[base + p + 256], 0, 1);   // global_prefetch_b8
        const float tmv = tmask[base + p];
        const int   l   = (tmv > 0.5f) ? (inst[base + p] & 7) : 0;
        const int   lk  = (kern[base + p] > 0.5f) ? l : 0;
        atomicAdd(&sCf[l], 1.0f);
        atomicAdd(&sCk[lk], 1.0f);
#pragma unroll
        for (int d = 0; d < Dch; ++d)
            atomicAdd(&sSum[lk * Dch + d], e0[(size_t)d * HWn + p]);
        if (l > 0) atomicMin(&sF[l], p);
    }
    __syncthreads();
    if (tid < Ln * Dch) atomicAdd(&F[OFF_SUMK + b * Ln * Dch + tid], sSum[tid]);
    if (tid < Ln) {
        atomicAdd(&F[OFF_CNTK + b * Ln + tid], sCk[tid]);
        atomicAdd(&F[OFF_CNTF + b * Ln + tid], sCf[tid]);
        atomicMin(&I[OFF_FIRST + b * Ln + tid], sF[tid]);
    }
}

// ---------------------------------------------------------------------------
// Pass 2: second occurrence = min index with label l, index != first[l].
__global__ __launch_bounds__(256) void k_second(const int*   __restrict__ inst,
                                                const float* __restrict__ tmask,
                                                float* F, int* I) {
    const int b = blockIdx.x / NB, chunk = blockIdx.x % NB;
    const int tid = threadIdx.x;
    (void)F;
    __shared__ int sS2[Ln];
    __shared__ int sF1[Ln];
    if (tid < Ln) { sS2[tid] = HWn; sF1[tid] = I[OFF_FIRST + b * Ln + tid]; }
    __syncthreads();

    const int base = b * HWn;
    const int start = chunk * CH;
    for (int p = start + tid; p < start + CH; p += 256) {
        const float tmv = tmask[base + p];
        const int   l   = (tmv > 0.5f) ? (inst[base + p] & 7) : 0;
        if (l > 0 && p != sF1[l]) atomicMin(&sS2[l], p);
    }
    __syncthreads();
    if (tid < Ln) atomicMin(&I[OFF_SECOND + b * Ln + tid], sS2[tid]);
}

// ---------------------------------------------------------------------------
// Per-image: means, coef_diag, Gram matrix of means via WMMA f32 16x16x4
// (M=16 labels padded, K=4 channels, N=16 labels; for a Gram product the A
// and B lane images coincide), then l_dis + l_reg. One wave per image,
// EXEC all-1s at the WMMA.
__global__ __launch_bounds__(32) void k_perimg(float* F, int* I) {
    const int b = blockIdx.x;
    const int lane = threadIdx.x;
    __shared__ float sM[16][Dch];       // padded label means (rows 8..15 = 0)
    __shared__ float sdG[Ln];           // squared norms of means
    __shared__ float sCd[Ln];
    __shared__ float sS[Ln], sT[Ln];
    __shared__ float sPart[32];
    const float diag = sqrtf((float)(Hn * Hn + Wn * Wn));

    if (lane < Ln) {
        const int l = lane;
        const float zm  = (l > 0) ? 1.0f : 0.0f;
        const float cnt = fmaxf(F[OFF_CNTK + b * Ln + l], 1.0f);
        float dg = 0.0f;
#pragma unroll
        for (int d = 0; d < Dch; ++d) {
            const float m = F[OFF_SUMK + (b * Ln + l) * Dch + d] / cnt * zm;
            sM[l][d] = m;
            F[OFF_MEAN + (b * Ln + l) * Dch + d] = m;
            dg += m * m;
        }
        sdG[l] = dg;
        int fi = I[OFF_FIRST  + b * Ln + l]; if (fi > HWn - 1) fi = HWn - 1;
        int se = I[OFF_SECOND + b * Ln + l]; if (se > HWn - 1) se = HWn - 1;
        const float r0 = (float)(fi / Wn), c0 = (float)(fi % Wn);
        const float r1 = (float)(se / Wn), c1 = (float)(se % Wn);
        const float cd = expf(sqrtf((r0 - c0) * (r0 - c0) +
                                    (r1 - c1) * (r1 - c1)) / diag * 0.5f) * zm;
        sCd[l] = cd;
        F[OFF_CD + b * Ln + l] = cd;
        sS[l] = r0 + c0;
        sT[l] = r1 + c1;
    } else if (lane < 16) {
#pragma unroll
        for (int d = 0; d < Dch; ++d) sM[lane][d] = 0.0f;
    }
    __syncthreads();

    // A layout 16x4: lane m = lane%16, k = vgpr + 2*(lane/16).
    // B layout 4x16: lane n = lane%16, k = vgpr + 2*(lane/16) -> B[k][n] = mean[n][k],
    // identical register image to A for a Gram product.
    const int m  = lane & 15;
    const int hi = lane >> 4;
    v2f a;
    a.x = sM[m][2 * hi];
    a.y = sM[m][2 * hi + 1];
    v8f c = {};
    c = __builtin_amdgcn_wmma_f32_16x16x4_f32(false, a, false, a,
                                              (short)0, c, false, false);
    // C/D layout: lane (hi=0) holds G[r][j] in c[r], j = lane%16.
    float part = 0.0f;
    if (hi == 0 && m >= 1 && m < 8) {
        const int j = m;
#pragma unroll
        for (int r = 1; r < 8; ++r) {
            if (r == j) continue;
            const float d2 = sdG[r] + sdG[j] - 2.0f * c[r];
            const float Dd = (d2 > 0.0f) ? sqrtf(d2) : 0.0f;
            const float ds = sS[r] - sS[j], dt = sT[r] - sT[j];
            const float dp = sqrtf(ds * ds + dt * dt);
            const float coef = 1.0f - 20.0f * expf(-4.0f - 2.5f * dp / diag);
            float v = 2.0f * DDc - coef * Dd;
            v = (v > 0.0f) ? v : 0.0f;
            part += logf(v * v + 1.0f);
        }
    }
    sPart[lane] = part;
    __syncthreads();
    if (lane == 0) {
        float ldis = 0.0f;
        for (int i = 0; i < 32; ++i) ldis += sPart[i];
        ldis /= (float)((Ln - 1) * (Ln - 2));
        float lreg = 0.0f;
        for (int l = 0; l < Ln; ++l) {
            const float n = (sdG[l] > 0.0f) ? sqrtf(sdG[l]) : 0.0f;
            lreg += logf(n + 1.0f);
        }
        lreg = lreg / (float)Ln * 0.001f;
        F[OFF_PER + b] = ldis + lreg;
    }
}

// ---------------------------------------------------------------------------
// Pass 3 (l_agg): per-pixel hinge against own-label mean, LDS-accumulated.
__global__ __launch_bounds__(256) void k_agg(const float* __restrict__ emb,
                                             const int*   __restrict__ inst,
                                             const float* __restrict__ tmask,
                                             float* F) {
    const int b = blockIdx.x / NB, chunk = blockIdx.x % NB;
    const int tid = threadIdx.x;
    __shared__ float sM[Ln][Dch];
    __shared__ float sCd[Ln];
    __shared__ float sV[Ln];
    if (tid < Ln * Dch) sM[tid / Dch][tid % Dch] = F[OFF_MEAN + b * Ln * Dch + tid];
    if (tid < Ln) { sCd[tid] = F[OFF_CD + b * Ln + tid]; sV[tid] = 0.0f; }
    __syncthreads();

    const int    base = b * HWn;
    const float* e0   = emb + (size_t)b * Dch * HWn;
    const int    start = chunk * CH;
    for (int p = start + tid; p < start + CH; p += 256) {
        __builtin_prefetch(&e0[p + 256], 0, 1);            // global_prefetch_b8
        const float tmv = tmask[base + p];
        const int   l   = (tmv > 0.5f) ? (inst[base + p] & 7) : 0;
        if (l > 0) {
            float sq = 0.0f;
#pragma unroll
            for (int d = 0; d < Dch; ++d) {
                const float df = e0[(size_t)d * HWn + p] - sM[l][d];
                sq += df * df;
            }
            const float dist = (sq > 0.0f) ? sqrtf(sq) : 0.0f;
            float v = sCd[l] * dist - DVc;
            v = (v > 0.0f) ? v : 0.0f;
            atomicAdd(&sV[l], logf(v * v + 1.0f));
        }
    }
    __syncthreads();
    if (tid < Ln) atomicAdd(&F[OFF_SUMV + b * Ln + tid], sV[tid]);
}

// ---------------------------------------------------------------------------
__global__ void k_final(const float* F, float* out) {
    if (blockIdx.x == 0 && threadIdx.x == 0) {
        float tot = 0.0f;
        for (int b = 0; b < Bn; ++b) {
            float lagg = 0.0f;
            for (int l = 1; l < Ln; ++l)
                lagg += F[OFF_SUMV + b * Ln + l] /
                        fmaxf(F[OFF_CNTF + b * Ln + l], 1.0f);
            lagg /= (float)(Ln - 1);
            tot += lagg + F[OFF_PER + b];
        }
        out[0] = tot / (float)Bn;   // LOSS_WEIGHT = 1
    }
}

// ---------------------------------------------------------------------------
extern "C" void kernel_launch(void* const* d_in, const int* in_sizes, int n_in,
                              void* d_out, int out_size, void* d_ws, size_t ws_size,
                              hipStream_t stream) {
    (void)in_sizes; (void)n_in; (void)out_size; (void)ws_size;
    const float* emb   = (const float*)d_in[0];
    const int*   inst  = (const int*)d_in[1];
    const float* kern  = (const float*)d_in[2];
    const float* tmask = (const float*)d_in[3];
    // d_in[4] = bboxes, unused by the reference.
    float* F = (float*)d_ws;
    int*   I = (int*)d_ws;
    float* out = (float*)d_out;

    k_init  <<<(WS_WORDS + 255) / 256, 256, 0, stream>>>(F, I);
    k_stats <<<Bn * NB, 256, 0, stream>>>(emb, inst, kern, tmask, F, I);
    k_second<<<Bn * NB, 256, 0, stream>>>(inst, tmask, F, I);
    k_perimg<<<Bn, 32, 0, stream>>>(F, I);
    k_agg   <<<Bn * NB, 256, 0, stream>>>(emb, inst, tmask, F);
    k_final <<<1, 32, 0, stream>>>(F, out);
}